// GINNet_33930241638747
// MI455X (gfx1250) — compile-verified
//
#include <hip/hip_runtime.h>

#define N_NODES   100000
#define N_EDGES   1200000
#define DIM       64
#define DIM_OUT   32
#define N_GRAPHS  256
#define BN_EPS    1e-5f

typedef float v2f __attribute__((ext_vector_type(2)));
typedef float v8f __attribute__((ext_vector_type(8)));

#define ZPAD 68   // zbuf row stride: (4*row + k) -> conflict-free, 8B aligned
#define WPAD 80   // wbuf row stride (float2): halves land on disjoint bank pairs

__device__ __forceinline__ v8f wmma4(v2f a, v2f b, v8f c) {
  // D = A(16x4 f32) x B(4x16 f32) + C(16x16 f32)
  return __builtin_amdgcn_wmma_f32_16x16x4_f32(
      /*neg_a=*/false, a, /*neg_b=*/false, b,
      /*c_mod=*/(short)0, c, /*reuse_a=*/false, /*reuse_b=*/false);
}

// Stage a 64x64 row-major weight matrix into LDS as paired rows:
// wbuf[p][c] = { W[2p][c], W[2p+1][c] }  -> one ds_load_b64 per B fragment.
__device__ __forceinline__ void stage_weights(const float* __restrict__ W,
                                              v2f (*__restrict__ wbuf)[WPAD],
                                              int tid) {
#pragma unroll
  for (int i = 0; i < 8; ++i) {
    const int idx = tid + i * 256;     // 2048 entries / 256 threads
    const int p   = idx >> 6;
    const int col = idx & 63;
    v2f w;
    w.x = W[(size_t)(2 * p) * DIM + col];
    w.y = W[(size_t)(2 * p + 1) * DIM + col];
    wbuf[p][col] = w;
  }
}

// ---------------------------------------------------------------- utilities
__global__ void zero_f32(float* __restrict__ p, int n) {
  int i = blockIdx.x * blockDim.x + threadIdx.x;
  if (i < n) p[i] = 0.0f;
}

// agg[dst] += feat[src]  (one thread per edge x float4 chunk)
__global__ void scatter_edges(const float* __restrict__ feat,
                              const long long* __restrict__ ei,
                              float* __restrict__ agg) {
  int t = blockIdx.x * blockDim.x + threadIdx.x;
  const int total = N_EDGES * 16;           // 16 float4 chunks per edge
  if (t >= total) return;
  int e = t >> 4;
  int c = (t & 15) << 2;
  long long s = ei[e];
  long long d = ei[N_EDGES + e];
  const float4 v = *(const float4*)(feat + (size_t)s * DIM + c);
  float* dp = agg + (size_t)d * DIM + c;
  atomicAdd(dp + 0, v.x);
  atomicAdd(dp + 1, v.y);
  atomicAdd(dp + 2, v.z);
  atomicAdd(dp + 3, v.w);
}

__global__ void count_nodes(const long long* __restrict__ batch,
                            float* __restrict__ cnt) {
  int i = blockIdx.x * blockDim.x + threadIdx.x;
  if (i < N_NODES) atomicAdd(&cnt[(int)batch[i]], 1.0f);
}

// --------------------------------------------------- fused GIN conv + BN
// One wave = one 16-node tile (8 waves/block). z = relu((x+agg)@Wa + ba);
// h = relu(z@Wb + bb); h = (h-m)*rsqrt(v+eps)*g + be.
// POOL: atomically pool into per-graph sums instead of storing h.
template <bool POOL>
__global__ __launch_bounds__(256) void gin_mlp(
    const float* __restrict__ xin, const float* __restrict__ agg,
    const float* __restrict__ Wa,  const float* __restrict__ ba,
    const float* __restrict__ Wb,  const float* __restrict__ bb,
    const float* __restrict__ gam, const float* __restrict__ bet,
    const float* __restrict__ mu,  const float* __restrict__ var,
    float* __restrict__ hout,
    const long long* __restrict__ batch, float* __restrict__ pool) {
  __shared__ v2f   wbuf[32][WPAD];           // paired weight rows (20.5 KB)
  __shared__ float zbuf[8][16][ZPAD];        // per-wave 16x64 intermediates

  const int tid   = threadIdx.x;
  const int lane  = tid & 31;
  const int wave  = tid >> 5;
  const int l15   = lane & 15;               // M for A-frag, N for B/C-frags
  const int kb    = (lane >> 4) << 1;        // K half-select: 0 or 2
  const int mhalf = (lane >> 4) << 3;        // C/D row half: 0 or 8
  const int tile  = blockIdx.x * 8 + wave;
  const int node  = tile * 16 + l15;
  const int ncl   = (node < N_NODES) ? node : 0;

  // ---- stage Wa ----
  stage_weights(Wa, wbuf, tid);
  __syncthreads();

  // ---- layer A: (x + agg) @ Wa ----
  v8f acc[4];
  {
    v8f zz = {};
    acc[0] = zz; acc[1] = zz; acc[2] = zz; acc[3] = zz;
  }
  const float* xp = xin + (size_t)ncl * DIM;
  const float* ap = agg + (size_t)ncl * DIM;
#pragma unroll
  for (int k0 = 0; k0 < DIM; k0 += 4) {
    const int k = k0 + kb;
    const int p = k >> 1;                    // paired-row index
    v2f a;
    a.x = xp[k]     + ap[k];
    a.y = xp[k + 1] + ap[k + 1];
#pragma unroll
    for (int nt = 0; nt < 4; ++nt)
      acc[nt] = wmma4(a, wbuf[p][l15 + nt * 16], acc[nt]);
  }
  // bias + ReLU -> LDS (row-major 16 x 64 per wave)
#pragma unroll
  for (int nt = 0; nt < 4; ++nt) {
    const int col  = l15 + nt * 16;
    const float bs = ba[col];
#pragma unroll
    for (int vv = 0; vv < 8; ++vv)
      zbuf[wave][vv + mhalf][col] = fmaxf(acc[nt][vv] + bs, 0.0f);
  }
  __syncthreads();                           // zbuf ready; wbuf free to re-stage

  // ---- stage Wb ----
  stage_weights(Wb, wbuf, tid);
  __syncthreads();

  // ---- layer B: z @ Wb ----
  {
    v8f zz = {};
    acc[0] = zz; acc[1] = zz; acc[2] = zz; acc[3] = zz;
  }
#pragma unroll
  for (int k0 = 0; k0 < DIM; k0 += 4) {
    const int k = k0 + kb;
    const int p = k >> 1;
    const v2f a = *(const v2f*)&zbuf[wave][l15][k];   // 8B aligned (68*4, k even)
#pragma unroll
    for (int nt = 0; nt < 4; ++nt)
      acc[nt] = wmma4(a, wbuf[p][l15 + nt * 16], acc[nt]);
  }
  // bias + ReLU + BN, then store or pool
#pragma unroll
  for (int nt = 0; nt < 4; ++nt) {
    const int col     = l15 + nt * 16;
    const float bs    = bb[col];
    const float scale = gam[col] * rsqrtf(var[col] + BN_EPS);
    const float shift = bet[col] - mu[col] * scale;
#pragma unroll
    for (int vv = 0; vv < 8; ++vv) {
      float val = fmaxf(acc[nt][vv] + bs, 0.0f);
      val = val * scale + shift;
      const int nr = tile * 16 + vv + mhalf;
      if (nr < N_NODES) {
        if (POOL) {
          const int bg = (int)batch[nr];
          atomicAdd(&pool[(size_t)bg * DIM + col], val);
        } else {
          hout[(size_t)nr * DIM + col] = val;
        }
      }
    }
  }
}

// ------------------------------------------------------------- final head
// out = relu((sums/cnt)@fw1 + fb1) @ fw2 + fb2 ;  grid = 2 blocks x 8 waves
__global__ __launch_bounds__(256) void head_kernel(
    const float* __restrict__ pool, const float* __restrict__ cnt,
    const float* __restrict__ fw1,  const float* __restrict__ fb1,
    const float* __restrict__ fw2,  const float* __restrict__ fb2,
    float* __restrict__ out) {
  __shared__ float zbuf[8][16][ZPAD];

  const int lane  = threadIdx.x & 31;
  const int wave  = threadIdx.x >> 5;
  const int l15   = lane & 15;
  const int kb    = (lane >> 4) << 1;
  const int mhalf = (lane >> 4) << 3;
  const int tile  = blockIdx.x * 8 + wave;   // 0..15
  const int row   = tile * 16 + l15;         // 0..255

  const float inv = 1.0f / fmaxf(cnt[row], 1.0f);
  const float* pr = pool + (size_t)row * DIM;

  v8f acc[4];
  {
    v8f zz = {};
    acc[0] = zz; acc[1] = zz; acc[2] = zz; acc[3] = zz;
  }
#pragma unroll
  for (int k0 = 0; k0 < DIM; k0 += 4) {
    const int k = k0 + kb;
    v2f a;
    a.x = pr[k] * inv;
    a.y = pr[k + 1] * inv;
    const float* w = fw1 + (size_t)k * DIM + l15;
#pragma unroll
    for (int nt = 0; nt < 4; ++nt) {
      v2f b;
      b.x = w[nt * 16];
      b.y = w[DIM + nt * 16];
      acc[nt] = wmma4(a, b, acc[nt]);
    }
  }
#pragma unroll
  for (int nt = 0; nt < 4; ++nt) {
    const int col  = l15 + nt * 16;
    const float bs = fb1[col];
#pragma unroll
    for (int vv = 0; vv < 8; ++vv)
      zbuf[wave][vv + mhalf][col] = fmaxf(acc[nt][vv] + bs, 0.0f);
  }
  __syncthreads();

  // second GEMM: 64 x 32 -> two N-tiles
  v8f o0, o1;
  {
    v8f zz = {};
    o0 = zz; o1 = zz;
  }
#pragma unroll
  for (int k0 = 0; k0 < DIM; k0 += 4) {
    const int k = k0 + kb;
    const v2f a = *(const v2f*)&zbuf[wave][l15][k];
    const float* w = fw2 + (size_t)k * DIM_OUT + l15;
    v2f b;
    b.x = w[0];
    b.y = w[DIM_OUT];
    o0 = wmma4(a, b, o0);
    b.x = w[16];
    b.y = w[DIM_OUT + 16];
    o1 = wmma4(a, b, o1);
  }
#pragma unroll
  for (int nt = 0; nt < 2; ++nt) {
    const int col  = l15 + nt * 16;
    const float bs = fb2[col];
    const v8f o    = nt ? o1 : o0;
#pragma unroll
    for (int vv = 0; vv < 8; ++vv) {
      const int r = tile * 16 + vv + mhalf;
      out[(size_t)r * DIM_OUT + col] = o[vv] + bs;
    }
  }
}

// --------------------------------------------------------------- launcher
extern "C" void kernel_launch(void* const* d_in, const int* in_sizes, int n_in,
                              void* d_out, int out_size, void* d_ws, size_t ws_size,
                              hipStream_t stream) {
  const float*     x     = (const float*)d_in[0];
  const long long* ei    = (const long long*)d_in[1];
  const long long* batch = (const long long*)d_in[2];
  const float* w1a = (const float*)d_in[3];
  const float* b1a = (const float*)d_in[4];
  const float* w1b = (const float*)d_in[5];
  const float* b1b = (const float*)d_in[6];
  const float* g1  = (const float*)d_in[7];
  const float* be1 = (const float*)d_in[8];
  const float* m1  = (const float*)d_in[9];
  const float* v1  = (const float*)d_in[10];
  const float* w2a = (const float*)d_in[11];
  const float* b2a = (const float*)d_in[12];
  const float* w2b = (const float*)d_in[13];
  const float* b2b = (const float*)d_in[14];
  const float* g2  = (const float*)d_in[15];
  const float* be2 = (const float*)d_in[16];
  const float* m2  = (const float*)d_in[17];
  const float* v2  = (const float*)d_in[18];
  const float* fw1 = (const float*)d_in[19];
  const float* fb1 = (const float*)d_in[20];
  const float* fw2 = (const float*)d_in[21];
  const float* fb2 = (const float*)d_in[22];
  float* out = (float*)d_out;

  // workspace layout
  float* agg  = (float*)d_ws;                    // N_NODES*DIM
  float* h1   = agg + (size_t)N_NODES * DIM;     // N_NODES*DIM
  float* pool = h1 + (size_t)N_NODES * DIM;      // N_GRAPHS*DIM
  float* cnt  = pool + (size_t)N_GRAPHS * DIM;   // N_GRAPHS

  const int nfeat  = N_NODES * DIM;
  const int sgrid  = (N_EDGES * 16 + 255) / 256;
  const int tiles  = (N_NODES + 15) / 16;
  const int mgrid  = (tiles + 7) / 8;

  // ---- GIN conv 1 ----
  zero_f32<<<(nfeat + 255) / 256, 256, 0, stream>>>(agg, nfeat);
  scatter_edges<<<sgrid, 256, 0, stream>>>(x, ei, agg);
  gin_mlp<false><<<mgrid, 256, 0, stream>>>(x, agg, w1a, b1a, w1b, b1b,
                                            g1, be1, m1, v1, h1, nullptr, nullptr);

  // ---- GIN conv 2 (+ fused graph mean-pool numerator) ----
  zero_f32<<<(nfeat + 255) / 256, 256, 0, stream>>>(agg, nfeat);
  zero_f32<<<(N_GRAPHS * DIM + N_GRAPHS + 255) / 256, 256, 0, stream>>>(
      pool, N_GRAPHS * DIM + N_GRAPHS);          // pool sums + cnt (contiguous)
  scatter_edges<<<sgrid, 256, 0, stream>>>(h1, ei, agg);
  count_nodes<<<(N_NODES + 255) / 256, 256, 0, stream>>>(batch, cnt);
  gin_mlp<true><<<mgrid, 256, 0, stream>>>(h1, agg, w2a, b2a, w2b, b2b,
                                           g2, be2, m2, v2, nullptr, batch, pool);

  // ---- head ----
  head_kernel<<<2, 256, 0, stream>>>(pool, cnt, fw1, fb1, fw2, fb2, out);
}